// NopadBaichuanAttention_47090021433670
// MI455X (gfx1250) — compile-verified
//
#include <hip/hip_runtime.h>
#include <hip/hip_bf16.h>
#include <math.h>

// ---------------------------------------------------------------------------
// Problem constants (from reference setup_inputs):
//   B=T=4, E=5120, H=40, D=128, BS=16, NB=512, MB=128, S=2048
// ---------------------------------------------------------------------------
#define TROWS  4
#define EDIM   5120
#define HEADS  40
#define HDIM   128
#define BLKSZ  16
#define MAXBLK 128

#define KSPLIT 4      // K-dimension split of the GEMMs (parallelism / in-flight)
#define ACHUNK 4      // key-range split of attention (blockIdx.z)

typedef __attribute__((ext_vector_type(2))) float v2f;
typedef __attribute__((ext_vector_type(8))) float v8f;

#if __has_builtin(__builtin_amdgcn_wmma_f32_16x16x4_f32)
#define USE_WMMA_F32X4 1
#else
#define USE_WMMA_F32X4 0
#endif

// ---------------------------------------------------------------------------
// GEMM partial:  Ypart[s][c][t][n] = sum_{k in split s} X[t][k] * W[c][k][n]
// K = N = EDIM are compile-time so every B load folds into ONE base address
// plus a 24-bit immediate offset; the weight pointer advances by 4 rows/iter.
// Wave = 16x32 output tile via two V_WMMA_F32_16X16X4_F32 accumulators.
// A (16x4 f32): lane&15 = M row; lanes 0-15 hold K=k,k+1; lanes 16-31 K=k+2,k+3.
// B (4x16 f32): col = n0+(lane&15); VGPR0 rows {k,k+2}, VGPR1 rows {k+1,k+3}.
// B fragments are software-pipelined one iteration ahead (steady-state loop has
// no guard); weight lines are prefetched 64 rows ahead (per-lane row
// k+64+(lane&3), col n0 touches all 4 upcoming 128B lines in one instruction).
// ---------------------------------------------------------------------------
#define KTILE    640                     // (EDIM/KSPLIT) = 1280 = 2*KTILE
#define GEMM_WPB 2                       // 64 threads, 64 output columns/block

__global__ __launch_bounds__(GEMM_WPB * 32)
void gemm_wmma_f32(const float* __restrict__ X,     // [TROWS, EDIM]
                   const float* __restrict__ Wt,    // [C, EDIM, EDIM]
                   float*       __restrict__ Ypart, // [KSPLIT, C, TROWS, EDIM]
                   int C)
{
  constexpr int K = EDIM;
  constexpr int N = EDIM;

  __shared__ __align__(16) float ldsA[TROWS * KTILE];

  const int tid  = threadIdx.x;
  const int lane = tid & 31;
  const int wave = tid >> 5;
  const int ks   = blockIdx.y;                    // K split index
  const int c    = blockIdx.z;

  const float* Wc = Wt + (size_t)c * K * N;
  float* Yc = Ypart + ((size_t)ks * C + c) * TROWS * N;

  const int n0 = (blockIdx.x * GEMM_WPB + wave) * 32;  // 32 cols per wave
  const int m  = lane & 15;            // A row / B column-within-tile
  const int kb = (lane >> 4) << 1;     // 0 (lanes 0-15) or 2 (lanes 16-31)

  const int kbeg = ks * (K / KSPLIT);
  const int kend = kbeg + (K / KSPLIT);

#if USE_WMMA_F32X4
  v8f acc0 = {};
  v8f acc1 = {};
#else
  float facc[TROWS] = {0.f, 0.f, 0.f, 0.f};
#endif

  for (int k0 = kbeg; k0 < kend; k0 += KTILE) {
    __syncthreads();
    // Stage A tile (TROWS x KTILE) into LDS, float4 per thread per step.
    for (int i = tid * 4; i < TROWS * KTILE; i += GEMM_WPB * 32 * 4) {
      int r  = i / KTILE;
      int kk = i - r * KTILE;
      *(float4*)&ldsA[i] = *(const float4*)&X[(size_t)r * K + k0 + kk];
    }
    __syncthreads();

#if USE_WMMA_F32X4
    const float* ldsArow = &ldsA[(m < TROWS ? m : 0) * KTILE];
    const bool   valid   = (m < TROWS);

    // Running pointers: wp = &W[k0+kb][n0+m], pf = prefetch 64 rows ahead.
    const float* wp = Wc + (size_t)(k0 + kb) * N + n0 + m;
    const float* pf = Wc + (size_t)(k0 + 64 + (lane & 3)) * N + n0;

    // Prime the pipeline for kk = 0 (immediate offsets: 0, N, 16, N+16).
    v2f b0, b1;
    b0.x = wp[0];
    b0.y = wp[N];
    b1.x = wp[16];
    b1.y = wp[N + 16];

    #pragma unroll 4
    for (int kk = 0; kk < KTILE - 4; kk += 4) {
      // Next iteration's B fragments: same base, +4 rows via immediates.
      v2f nb0, nb1;
      nb0.x = wp[4 * N];
      nb0.y = wp[5 * N];
      nb1.x = wp[4 * N + 16];
      nb1.y = wp[5 * N + 16];

      __builtin_prefetch(pf, 0, 0);     // 4 upcoming rows' 128B lines

      // A fragment: two consecutive K values of this lane's row from LDS.
      v2f av = *(const v2f*)&ldsArow[kk + kb];
      v2f a;
      a.x = valid ? av.x : 0.0f;
      a.y = valid ? av.y : 0.0f;

      acc0 = __builtin_amdgcn_wmma_f32_16x16x4_f32(false, a, false, b0,
                                                   (short)0, acc0, false, false);
      acc1 = __builtin_amdgcn_wmma_f32_16x16x4_f32(false, a, false, b1,
                                                   (short)0, acc1, false, false);
      b0 = nb0;
      b1 = nb1;
      wp += 4 * (size_t)N;
      pf += 4 * (size_t)N;
    }
    {
      // Peeled last iteration (kk = KTILE-4): no further loads.
      v2f av = *(const v2f*)&ldsArow[(KTILE - 4) + kb];
      v2f a;
      a.x = valid ? av.x : 0.0f;
      a.y = valid ? av.y : 0.0f;
      acc0 = __builtin_amdgcn_wmma_f32_16x16x4_f32(false, a, false, b0,
                                                   (short)0, acc0, false, false);
      acc1 = __builtin_amdgcn_wmma_f32_16x16x4_f32(false, a, false, b1,
                                                   (short)0, acc1, false, false);
    }
#else
    for (int kk = 0; kk < KTILE; ++kk) {
      float w = Wc[(size_t)(k0 + kk) * N + n0 + lane];
      #pragma unroll
      for (int r = 0; r < TROWS; ++r)
        facc[r] = fmaf(ldsA[r * KTILE + kk], w, facc[r]);
    }
#endif
  }

#if USE_WMMA_F32X4
  // C/D layout: VGPR r holds M=r in lanes 0-15 (only rows 0..3 are real).
  if (lane < 16) {
    #pragma unroll
    for (int r = 0; r < TROWS; ++r) {
      Yc[(size_t)r * N + n0 + lane]      = acc0[r];
      Yc[(size_t)r * N + n0 + 16 + lane] = acc1[r];
    }
  }
#else
  #pragma unroll
  for (int r = 0; r < TROWS; ++r)
    Yc[(size_t)r * N + n0 + lane] = facc[r];
#endif
}

// Deterministic reduction of the KSPLIT partial GEMM outputs.
__global__ __launch_bounds__(256)
void gemm_merge(const float* __restrict__ part,  // [KSPLIT, n]
                float* __restrict__ out, int n)
{
  int i = blockIdx.x * 256 + threadIdx.x;
  if (i < n) {
    float s = part[i];
    #pragma unroll
    for (int z = 1; z < KSPLIT; ++z) s += part[(size_t)z * n + i];
    out[i] = s;
  }
}

// ---------------------------------------------------------------------------
// Flash-decode attention with ALiBi over the paged KV cache.
// Grid (H, B, ACHUNK); 8 waves/block. Virtual wave vw = chunk*8 + wave owns
// keys t = vw, vw+32, ... ; two keys are processed per loop iteration with all
// four K/V float4 loads issued before either key is consumed (latency hiding).
// Each block emits an online-softmax partial (m, l, unnormalized acc[128]).
// The new token's K/V comes from the QKV projection (inputs never mutated).
// ---------------------------------------------------------------------------
__global__ __launch_bounds__(256)
void attn_decode(const float* __restrict__ qkv,   // [3, TROWS, EDIM]
                 const float* __restrict__ kc,    // [NB, H, BS, D]
                 const float* __restrict__ vc,
                 const int*   __restrict__ bt,    // [TROWS, MAXBLK]
                 const int*   __restrict__ slen,  // [TROWS]
                 float* __restrict__ pm,          // [ACHUNK, TROWS*HEADS]
                 float* __restrict__ pl,          // [ACHUNK, TROWS*HEADS]
                 float* __restrict__ pacc)        // [ACHUNK, TROWS*HEADS, HDIM]
{
  const int h     = blockIdx.x;
  const int b     = blockIdx.y;
  const int chunk = blockIdx.z;
  const int tid   = threadIdx.x;
  const int lane  = tid & 31;
  const int wave  = tid >> 5;                 // 0..7
  const int vw    = chunk * 8 + wave;         // 0..31, key stride 32

  const int Lq  = slen[b];
  const int pos = Lq - 1;

  const float* q  = qkv + (size_t)b * EDIM + h * HDIM;
  const float* kn = qkv + (size_t)1 * TROWS * EDIM + (size_t)b * EDIM + h * HDIM;
  const float* vn = qkv + (size_t)2 * TROWS * EDIM + (size_t)b * EDIM + h * HDIM;

  const float4 q4 = *(const float4*)(q + lane * 4);
  const float  sm_scale = 0.088388347648318447f;  // 1/sqrt(128)
  // ALiBi slopes for H=40.
  const float slope = (h < 32) ? exp2f(-0.25f * (float)(h + 1))
                               : exp2f(-0.125f * (float)(2 * (h - 32) + 1));

  float  mx  = -__builtin_inff();
  float  lse = 0.0f;
  float4 acc = make_float4(0.f, 0.f, 0.f, 0.f);

  const int* btb = bt + b * MAXBLK;
  const int  hoff = h * BLKSZ * HDIM + lane * 4;   // per-head, per-lane offset

  // Returns byte-folded element offset into kc/vc for key t (t != pos).
  auto cache_off = [&](int t) -> size_t {
    int blk = btb[t >> 4];
    return (size_t)blk * (HEADS * BLKSZ * HDIM) + hoff + (t & (BLKSZ - 1)) * HDIM;
  };
  auto online = [&](int t, const float4& k4, const float4& v4) {
    float d = q4.x * k4.x + q4.y * k4.y + q4.z * k4.z + q4.w * k4.w;
    #pragma unroll
    for (int s = 16; s > 0; s >>= 1) d += __shfl_xor(d, s, 32);
    float score = d * sm_scale + slope * (float)(t - pos);
    float mn    = fmaxf(mx, score);
    float corr  = __expf(mx - mn);
    float p     = __expf(score - mn);
    lse   = lse * corr + p;
    acc.x = acc.x * corr + p * v4.x;
    acc.y = acc.y * corr + p * v4.y;
    acc.z = acc.z * corr + p * v4.z;
    acc.w = acc.w * corr + p * v4.w;
    mx = mn;
  };

  for (int t = vw; t < Lq; t += 64) {
    const int t1 = t + 32;

    float4 k40, v40;
    if (t == pos) {
      k40 = *(const float4*)(kn + lane * 4);
      v40 = *(const float4*)(vn + lane * 4);
    } else {
      size_t o = cache_off(t);
      k40 = *(const float4*)(kc + o);
      v40 = *(const float4*)(vc + o);
    }

    float4 k41, v41;
    const bool have1 = (t1 < Lq);
    if (have1) {                                // wave-uniform branch
      if (t1 == pos) {
        k41 = *(const float4*)(kn + lane * 4);
        v41 = *(const float4*)(vn + lane * 4);
      } else {
        size_t o = cache_off(t1);
        k41 = *(const float4*)(kc + o);
        v41 = *(const float4*)(vc + o);
      }
    }

    online(t, k40, v40);
    if (have1) online(t1, k41, v41);
  }

  // Merge the 8 waves of this chunk through LDS into one partial.
  __shared__ float sm_m[8];
  __shared__ float sm_l[8];
  __shared__ __align__(16) float sm_acc[8][HDIM];
  if (lane == 0) { sm_m[wave] = mx; sm_l[wave] = lse; }
  *(float4*)&sm_acc[wave][lane * 4] = acc;
  __syncthreads();

  const int bh = b * HEADS + h;
  if (tid < HDIM) {
    float M = -__builtin_inff();
    #pragma unroll
    for (int w = 0; w < 8; ++w)
      if (sm_l[w] > 0.f) M = fmaxf(M, sm_m[w]);
    float Ls = 0.f, s = 0.f;
    #pragma unroll
    for (int w = 0; w < 8; ++w) {
      if (sm_l[w] > 0.f) {
        float f = __expf(sm_m[w] - M);
        Ls += sm_l[w] * f;
        s  += sm_acc[w][tid] * f;
      }
    }
    pacc[((size_t)chunk * TROWS * HEADS + bh) * HDIM + tid] = s;
    if (tid == 0) {
      pm[(size_t)chunk * TROWS * HEADS + bh] = M;
      pl[(size_t)chunk * TROWS * HEADS + bh] = Ls;
    }
  }
}

// Combine the ACHUNK per-chunk softmax partials into the attention output.
__global__ __launch_bounds__(HDIM)
void attn_merge(const float* __restrict__ pm, const float* __restrict__ pl,
                const float* __restrict__ pacc,
                float* __restrict__ out)         // [TROWS, EDIM]
{
  const int h  = blockIdx.x;
  const int b  = blockIdx.y;
  const int d  = threadIdx.x;
  const int bh = b * HEADS + h;
  const int NP = TROWS * HEADS;

  float M = -__builtin_inff();
  #pragma unroll
  for (int z = 0; z < ACHUNK; ++z)
    if (pl[(size_t)z * NP + bh] > 0.f) M = fmaxf(M, pm[(size_t)z * NP + bh]);

  float Ls = 0.f, s = 0.f;
  #pragma unroll
  for (int z = 0; z < ACHUNK; ++z) {
    float lz = pl[(size_t)z * NP + bh];
    if (lz > 0.f) {
      float f = __expf(pm[(size_t)z * NP + bh] - M);
      Ls += lz * f;
      s  += pacc[((size_t)z * NP + bh) * HDIM + d] * f;
    }
  }
  out[(size_t)b * EDIM + h * HDIM + d] = s / Ls;
}

// ---------------------------------------------------------------------------
// Launch. Workspace layout (floats):
//   qkv_part [KSPLIT*3*4*E] | qkv [3*4*E] | pm [ACHUNK*160] | pl [ACHUNK*160]
//   | pacc [ACHUNK*160*128] | attn_out [4*E] | oproj_part [KSPLIT*4*E]
// Total ~1.93 MB.
// ---------------------------------------------------------------------------
extern "C" void kernel_launch(void* const* d_in, const int* in_sizes, int n_in,
                              void* d_out, int out_size, void* d_ws, size_t ws_size,
                              hipStream_t stream)
{
  const float* hidden = (const float*)d_in[0];   // [4, 5120]
  const float* qkv_w  = (const float*)d_in[1];   // [3, 5120, 5120]
  const float* o_w    = (const float*)d_in[2];   // [5120, 5120]
  const float* kc     = (const float*)d_in[3];   // [512, 40, 16, 128]
  const float* vc     = (const float*)d_in[4];
  const int*   bt     = (const int*)d_in[5];     // [4, 128]
  const int*   slen   = (const int*)d_in[6];     // [4]
  float*       out    = (float*)d_out;           // [4, 5120]

  const size_t NQKV = (size_t)3 * TROWS * EDIM;          // 61440
  const size_t NOUT = (size_t)TROWS * EDIM;              // 20480
  const size_t NP   = (size_t)TROWS * HEADS;             // 160

  float* qkv_part   = (float*)d_ws;
  float* qkv_ws     = qkv_part + (size_t)KSPLIT * NQKV;
  float* pm         = qkv_ws + NQKV;
  float* pl         = pm + (size_t)ACHUNK * NP;
  float* pacc       = pl + (size_t)ACHUNK * NP;
  float* attn_ws    = pacc + (size_t)ACHUNK * NP * HDIM;
  float* oproj_part = attn_ws + NOUT;

  // 1) QKV projection partials: 80 x KSPLIT x 3 = 960 blocks (1920 waves).
  dim3 gq(EDIM / (32 * GEMM_WPB), KSPLIT, 3);
  gemm_wmma_f32<<<gq, GEMM_WPB * 32, 0, stream>>>(hidden, qkv_w, qkv_part, 3);
  gemm_merge<<<(int)((NQKV + 255) / 256), 256, 0, stream>>>(qkv_part, qkv_ws,
                                                            (int)NQKV);

  // 2) Attention partials over ACHUNK key ranges, then merge.
  attn_decode<<<dim3(HEADS, TROWS, ACHUNK), 256, 0, stream>>>(
      qkv_ws, kc, vc, bt, slen, pm, pl, pacc);
  attn_merge<<<dim3(HEADS, TROWS), HDIM, 0, stream>>>(pm, pl, pacc, attn_ws);

  // 3) Output projection partials + merge.
  dim3 go(EDIM / (32 * GEMM_WPB), KSPLIT, 1);
  gemm_wmma_f32<<<go, GEMM_WPB * 32, 0, stream>>>(attn_ws, o_w, oproj_part, 1);
  gemm_merge<<<(int)((NOUT + 255) / 256), 256, 0, stream>>>(oproj_part, out,
                                                            (int)NOUT);
}